// GCNEncoder_65000035058237
// MI455X (gfx1250) — compile-verified
//
#include <hip/hip_runtime.h>

// ---------------------------------------------------------------------------
// GCN 2-layer forward for MI455X (gfx1250, wave32).
//   h1  = X @ W1                          (WMMA f32 16x16x4, full fp32)
//   o1  = Dhat (A+I) Dhat h1              (atomic scatter-add over edges)
//   h2  = relu(o1 + b1) @ W2              (bias+ReLU fused into GEMM2 A-load)
//   out = Dhat (A+I) Dhat h2 + b2         (atomic scatter-add over edges)
// Aggregation working set (~70 MB) fits in the 192 MB L2, so the f32 atomics
// resolve at L2 atomic units; this is the bandwidth-dominant phase (~2.5 GB
// of L2 traffic -> O(100us) at MI455X bandwidths). GEMMs are <1 GFLOP total.
// ---------------------------------------------------------------------------

typedef __attribute__((ext_vector_type(2))) float v2f;
typedef __attribute__((ext_vector_type(8))) float v8f;

#define N_NODES 100000
#define IN_CH   128
#define HID_CH  16
#define OUT_CH  64
#define N_TILES ((N_NODES + 15) / 16)   // 6250 (exact)

// ---------------- degree / normalization ----------------------------------

__global__ void k_deg_init(float* __restrict__ deg) {
  int i = blockIdx.x * blockDim.x + threadIdx.x;
  if (i < N_NODES) deg[i] = 1.0f;            // self-loop contributes 1
}

__global__ void k_degree(const int* __restrict__ col, float* __restrict__ deg, int E) {
  int e = blockIdx.x * blockDim.x + threadIdx.x;
  if (e < E) atomicAdd(&deg[col[e]], 1.0f);
}

__global__ void k_dinv(float* __restrict__ deg) {
  int i = blockIdx.x * blockDim.x + threadIdx.x;
  if (i < N_NODES) deg[i] = rsqrtf(deg[i]);  // deg >= 1 always (self loops)
}

// ---------------- GEMM1: h1[N,16] = x[N,128] @ W1[128,16] ------------------
// One wave32 per 16x16 output tile; K in steps of 4 via V_WMMA_F32_16X16X4_F32.

__global__ __launch_bounds__(256) void k_gemm1(const float* __restrict__ x,
                                               const float* __restrict__ W1,
                                               float* __restrict__ h1) {
  const int tile = blockIdx.x * 8 + (threadIdx.x >> 5);   // row tile (wave-uniform)
  if (tile >= N_TILES) return;                            // whole-wave guard: EXEC all-1s
  const int lane = threadIdx.x & 31;
  const int half = lane >> 4;                             // 0: K pair {0,1}; 1: K pair {2,3}
  const int l16  = lane & 15;
  const int row  = tile * 16 + l16;                       // A row owned by this lane

  v8f c = {};
  #pragma unroll 4
  for (int k = 0; k < IN_CH; k += 4) {
    // A fragment: 16x4 f32 -> 2 VGPRs/lane (ISA 7.12.2 layout)
    v2f a = *(const v2f*)(x + (size_t)row * IN_CH + k + half * 2);
    // B fragment: 4x16 f32, rows striped across lanes, K pair by lane half
    v2f b;
    b.x = W1[(k + half * 2 + 0) * HID_CH + l16];
    b.y = W1[(k + half * 2 + 1) * HID_CH + l16];
    c = __builtin_amdgcn_wmma_f32_16x16x4_f32(false, a, false, b,
                                              (short)0, c, false, false);
  }
  // C/D layout: VGPR v -> M = v + half*8, N = l16
  #pragma unroll
  for (int v = 0; v < 8; ++v)
    h1[(size_t)(tile * 16 + v + half * 8) * HID_CH + l16] = c[v];
}

// ---------------- layer-1 aggregation --------------------------------------

// out1[i,f] = h1[i,f] * dinv[i]^2   (self-loop term, also initializes out1)
__global__ void k_self_init16(const float* __restrict__ h1,
                              const float* __restrict__ dinv,
                              float* __restrict__ out1) {
  int t = blockIdx.x * blockDim.x + threadIdx.x;
  if (t < N_NODES * HID_CH) {
    int i = t >> 4;
    float d = dinv[i];
    out1[t] = h1[t] * d * d;
  }
}

// edge scatter: out1[col,f] += h1[row,f] * dinv[row]*dinv[col]
// 16 lanes per edge: gathers are 64B-contiguous per edge, atomics cacheline-
// contiguous -> coalesced L2 atomic traffic.
__global__ void k_agg16(const int* __restrict__ row, const int* __restrict__ col,
                        const float* __restrict__ h1, const float* __restrict__ dinv,
                        float* __restrict__ out1, int E) {
  int t = blockIdx.x * blockDim.x + threadIdx.x;   // E*16 = 51.2M < 2^31
  if (t >= E * HID_CH) return;
  int e = t >> 4;
  int f = t & 15;
  int r = row[e];                                  // same-address within edge group
  int c = col[e];
  float nrm = dinv[r] * dinv[c];
  atomicAdd(&out1[c * HID_CH + f], h1[(size_t)r * HID_CH + f] * nrm);
}

// ---------------- GEMM2: h2[N,64] = relu(out1 + b1) @ W2[16,64] ------------
// bias + ReLU fused into the A-fragment load (K index == hidden feature),
// saving a full elementwise pass over the 6.4 MB hidden matrix.
// blockIdx.y = column tile (4 tiles of 16); K = 16 -> 4 WMMA steps.

__global__ __launch_bounds__(256) void k_gemm2(const float* __restrict__ h,
                                               const float* __restrict__ b1,
                                               const float* __restrict__ W2,
                                               float* __restrict__ h2) {
  const int tile = blockIdx.x * 8 + (threadIdx.x >> 5);
  if (tile >= N_TILES) return;
  const int ct   = blockIdx.y;                     // 0..3
  const int lane = threadIdx.x & 31;
  const int half = lane >> 4;
  const int l16  = lane & 15;
  const int row  = tile * 16 + l16;

  v8f c = {};
  #pragma unroll
  for (int k = 0; k < HID_CH; k += 4) {
    const int k0 = k + half * 2;
    v2f a = *(const v2f*)(h + (size_t)row * HID_CH + k0);
    a.x = fmaxf(a.x + b1[k0 + 0], 0.0f);           // fused bias + ReLU
    a.y = fmaxf(a.y + b1[k0 + 1], 0.0f);
    v2f b;
    b.x = W2[(k0 + 0) * OUT_CH + ct * 16 + l16];
    b.y = W2[(k0 + 1) * OUT_CH + ct * 16 + l16];
    c = __builtin_amdgcn_wmma_f32_16x16x4_f32(false, a, false, b,
                                              (short)0, c, false, false);
  }
  #pragma unroll
  for (int v = 0; v < 8; ++v)
    h2[(size_t)(tile * 16 + v + half * 8) * OUT_CH + ct * 16 + l16] = c[v];
}

// ---------------- layer-2 aggregation --------------------------------------

// out[i,f] = h2[i,f] * dinv[i]^2 + b2[f]   (self-loop + bias, initializes d_out)
__global__ void k_self_init64(const float* __restrict__ h2,
                              const float* __restrict__ dinv,
                              const float* __restrict__ b2,
                              float* __restrict__ out) {
  int t = blockIdx.x * blockDim.x + threadIdx.x;
  if (t < N_NODES * OUT_CH) {
    int i = t >> 6;
    int f = t & 63;
    float d = dinv[i];
    out[t] = h2[t] * d * d + b2[f];
  }
}

// edge scatter: out[col,f] += h2[row,f] * dinv[row]*dinv[col]
// 64 lanes (2 waves) per edge: 256B-contiguous gathers and atomics.
__global__ void k_agg64(const int* __restrict__ row, const int* __restrict__ col,
                        const float* __restrict__ h2, const float* __restrict__ dinv,
                        float* __restrict__ out, int E) {
  int t = blockIdx.x * blockDim.x + threadIdx.x;   // E*64 = 204.8M < 2^31
  if (t >= E * OUT_CH) return;
  int e = t >> 6;
  int f = t & 63;
  int r = row[e];
  int c = col[e];
  float nrm = dinv[r] * dinv[c];
  atomicAdd(&out[c * OUT_CH + f], h2[(size_t)r * OUT_CH + f] * nrm);
}

// ---------------------------------------------------------------------------

extern "C" void kernel_launch(void* const* d_in, const int* in_sizes, int n_in,
                              void* d_out, int out_size, void* d_ws, size_t ws_size,
                              hipStream_t stream) {
  const float* x   = (const float*)d_in[0];
  const int*   ei  = (const int*)d_in[1];      // [2, E] flat, int32
  const float* W1  = (const float*)d_in[2];
  const float* b1  = (const float*)d_in[3];
  const float* W2  = (const float*)d_in[4];
  const float* b2  = (const float*)d_in[5];
  float*       out = (float*)d_out;

  const int E = in_sizes[1] / 2;
  const int* row = ei;        // sources
  const int* col = ei + E;    // targets (aggregation index)

  // workspace carve-up (floats): dinv | h1 | out1 | h2   (~39 MB total)
  float* dinv = (float*)d_ws;
  float* h1   = dinv + N_NODES;
  float* out1 = h1   + (size_t)N_NODES * HID_CH;
  float* h2   = out1 + (size_t)N_NODES * HID_CH;

  const int B = 256;

  // normalization coefficients
  k_deg_init<<<(N_NODES + B - 1) / B, B, 0, stream>>>(dinv);
  k_degree  <<<(E + B - 1) / B, B, 0, stream>>>(col, dinv, E);
  k_dinv    <<<(N_NODES + B - 1) / B, B, 0, stream>>>(dinv);

  // layer 1
  k_gemm1<<<(N_TILES + 7) / 8, B, 0, stream>>>(x, W1, h1);
  k_self_init16<<<(N_NODES * HID_CH + B - 1) / B, B, 0, stream>>>(h1, dinv, out1);
  k_agg16<<<(E * HID_CH + B - 1) / B, B, 0, stream>>>(row, col, h1, dinv, out1, E);

  // layer 2 (bias+ReLU fused into GEMM2 A-load)
  dim3 g2((N_TILES + 7) / 8, OUT_CH / 16);
  k_gemm2<<<g2, B, 0, stream>>>(out1, b1, W2, h2);
  k_self_init64<<<(N_NODES * OUT_CH + B - 1) / B, B, 0, stream>>>(h2, dinv, b2, out);
  k_agg64<<<(E * OUT_CH + B - 1) / B, B, 0, stream>>>(row, col, h2, dinv, out, E);
}